// ACTLTC_43542378447320
// MI455X (gfx1250) — compile-verified
//
#include <hip/hip_runtime.h>
#include <math.h>

#define B_   256
#define T_   128
#define S_   128
#define U_   256
#define UNF  6
#define MAXC 10
#define EPS_ACT 0.01f
#define EPS_DIV 1e-8f
#define LOG2E   1.4426950408889634f

typedef float v2f __attribute__((ext_vector_type(2)));
typedef float v8f __attribute__((ext_vector_type(8)));

__device__ __forceinline__ float fast_exp2(float x) { return __builtin_amdgcn_exp2f(x); }
__device__ __forceinline__ float fast_rcp(float x)  { return __builtin_amdgcn_rcpf(x); }
// sigmoid(sigma*(v-mu)) with a = sigma*log2e, b = sigma*mu*log2e prefolded
__device__ __forceinline__ float sigm_ab(float a, float b, float v) {
    return fast_rcp(1.0f + fast_exp2(b - a * v));
}
__device__ __forceinline__ float softplus_f(float x) { return logf(1.0f + expf(x)); }

// Pack per-edge params: {sigma*log2e, sigma*mu*log2e, softplus(w)*erev, softplus(w)}
__global__ void actltc_prep(const float* __restrict__ w, const float* __restrict__ mu,
                            const float* __restrict__ sg, const float* __restrict__ er,
                            float4* __restrict__ out, int n) {
    int e = blockIdx.x * blockDim.x + threadIdx.x;
    if (e >= n) return;
    float a  = sg[e] * LOG2E;
    float b  = sg[e] * mu[e] * LOG2E;
    float wp = softplus_f(w[e]);
    out[e] = make_float4(a, b, wp * er[e], wp);
}

__launch_bounds__(512)
__global__ void actltc_main(const float4* __restrict__ pk,   // [U_*U_] edge params
                            const float4* __restrict__ sk,   // [S_*U_] sensory params
                            const float*  __restrict__ x,    // [B,T,S]
                            const float*  __restrict__ iw, const float* __restrict__ ib,
                            const float*  __restrict__ gleak, const float* __restrict__ vleak,
                            const float*  __restrict__ cmp,
                            const float*  __restrict__ ow, const float* __restrict__ ob,
                            const float*  __restrict__ hw, const float* __restrict__ hb,
                            float* __restrict__ out_seq,     // [B,T,U]
                            float* __restrict__ out_state,   // [B,U]
                            float* __restrict__ ponder_part) // [B]
{
    __shared__ __align__(16) float vsh[U_];
    __shared__ float accsh[U_], wnum[U_], wden[U_], xns[U_], xds[U_];
    __shared__ float xinsh[S_];
    __shared__ float red[512];
    __shared__ float spn[512], spd[512];
    __shared__ float sWeight;

    const int tid  = (int)threadIdx.x;
    const int b    = (int)blockIdx.x;
    const int lane = tid & 31;
    const int wav  = tid >> 5;          // 0..15
    const int j0   = wav * 16;
    const int nl   = lane & 15;
    const int ih   = lane >> 4;         // 0 or 1
    const int jm   = j0 + nl;           // this lane's post-unit column

    // per-unit constants (threads 0..255 own unit j = tid)
    float cmt = 0.f, gl = 0.f, glv = 0.f, owj = 0.f, obj = 0.f, hwj = 0.f;
    float iwj = 0.f, ibj = 0.f;
    if (tid < U_) {
        gl  = softplus_f(gleak[tid]);
        cmt = softplus_f(cmp[tid]) * (float)UNF;   // cm_t = softplus(cm)/(1/UNF)
        glv = gl * vleak[tid];
        owj = ow[tid]; obj = ob[tid]; hwj = hw[tid];
        vsh[tid] = 0.0f;
    }
    if (tid < S_) { iwj = iw[tid]; ibj = ib[tid]; }
    const float hb0 = hb[0];
    __syncthreads();

    const v2f onesA = {1.0f, 1.0f};
    const float4* __restrict__ colp = pk + jm;   // this lane's column base
    float ponder_total = 0.0f;                   // thread 0 only

    for (int t = 0; t < T_; ++t) {
        // affine input mapping into LDS
        if (tid < S_) xinsh[tid] = x[((size_t)b * T_ + t) * S_ + tid] * iwj + ibj;
        if (tid < U_) accsh[tid] = 0.0f;
        __syncthreads();

        // sensory synapse sums (computed once per t; reused by every ODE unfold)
        {
            int j  = tid & (U_ - 1);
            int s0 = (tid >> 8) * (S_ / 2);
            float pn = 0.f, pd = 0.f;
            for (int k = 0; k < S_ / 2; ++k) {
                int s = s0 + k;
                float4 q = sk[s * U_ + j];
                float sg = sigm_ab(q.x, q.y, xinsh[s]);
                pn = fmaf(q.z, sg, pn);
                pd = fmaf(q.w, sg, pd);
            }
            spn[tid] = pn; spd[tid] = pd;
        }
        __syncthreads();
        if (tid < U_) { xns[tid] = spn[tid] + spn[tid + U_]; xds[tid] = spd[tid] + spd[tid + U_]; }
        __syncthreads();

        // ACT scalars live in thread-0 registers
        float halt_sum = 0.f, rem = 0.f, nup = 0.f;
        int still = 1;

        for (int n = 0; n < MAXC; ++n) {
            for (int u = 0; u < UNF; ++u) {
                v8f Dn = {0,0,0,0,0,0,0,0};
                v8f Dd = {0,0,0,0,0,0,0,0};
                // software-pipelined: params for chunk i0 pre-loaded one trip ahead
                float4 p0 = colp[(0 + ih) * U_];
                float4 p1 = colp[(2 + ih) * U_];
                #pragma unroll 2
                for (int i0 = 0; i0 < U_; i0 += 4) {
                    float4 c0 = p0, c1 = p1;
                    const int inx = (i0 + 4) & (U_ - 1);        // branchless wrap
                    p0 = colp[(inx + ih) * U_];
                    p1 = colp[(inx + 2 + ih) * U_];
                    __builtin_prefetch(colp + (((i0 + 8) & (U_ - 1)) + ih) * U_, 0, 1);
                    float4 vv = *(const float4*)(&vsh[i0]);
                    float va = ih ? vv.y : vv.x;
                    float vb = ih ? vv.w : vv.z;
                    float s0v = sigm_ab(c0.x, c0.y, va);
                    float s1v = sigm_ab(c1.x, c1.y, vb);
                    v2f Bn = { c0.z * s0v, c1.z * s1v };
                    v2f Bd = { c0.w * s0v, c1.w * s1v };
                    // D[m,n] = sum_k B[k,n] + C  (A = ones): matrix-pipe reduction over 4 pre-units
                    Dn = __builtin_amdgcn_wmma_f32_16x16x4_f32(false, onesA, false, Bn,
                                                               (short)0, Dn, false, false);
                    Dd = __builtin_amdgcn_wmma_f32_16x16x4_f32(false, onesA, false, Bd,
                                                               (short)0, Dd, false, false);
                }
                if (lane < 16) { wnum[jm] = Dn[0]; wden[jm] = Dd[0]; }
                __syncthreads();
                if (tid < U_) {
                    float v    = vsh[tid];
                    float numv = wnum[tid] + xns[tid] + glv;
                    float denv = wden[tid] + xds[tid] + cmt + gl + EPS_DIV;
                    vsh[tid] = (cmt * v + numv) / denv;
                }
                __syncthreads();
            }
            // halting probability: p = sigmoid(v . halt_w + halt_b)
            red[tid] = (tid < U_) ? vsh[tid] * hwj : 0.0f;
            __syncthreads();
            for (int off = 256; off >= 1; off >>= 1) {
                if (tid < off) red[tid] += red[tid + off];
                __syncthreads();
            }
            if (tid == 0) {
                float logit  = red[0] + hb0;
                float p      = fast_rcp(1.0f + fast_exp2(-logit * LOG2E));
                float p_eff  = still ? p : 0.0f;
                float newsum = halt_sum + p_eff;
                int halting  = (n == MAXC - 1) ? still
                                               : (still && (newsum >= 1.0f - EPS_ACT));
                float r  = 1.0f - halt_sum;
                sWeight  = halting ? r : p_eff;
                rem     += halting ? r : 0.0f;
                nup     += still ? 1.0f : 0.0f;
                halt_sum = newsum;
                still    = still && !halting;
            }
            __syncthreads();
            if (tid < U_) accsh[tid] = fmaf(sWeight, vsh[tid], accsh[tid]);
            __syncthreads();
        }

        if (tid < U_) {
            float ns = accsh[tid];
            out_seq[((size_t)b * T_ + t) * U_ + tid] = ns * owj + obj;
            vsh[tid] = ns;                 // carried state = weighted ACT accumulation
        }
        if (tid == 0) ponder_total += nup + rem;
        __syncthreads();
    }

    if (tid < U_) out_state[(size_t)b * U_ + tid] = vsh[tid];
    if (tid == 0) ponder_part[b] = ponder_total;
}

__global__ void actltc_finalize(const float* __restrict__ part, float* __restrict__ out) {
    if (blockIdx.x == 0 && threadIdx.x == 0) {
        float s = 0.f;
        for (int i = 0; i < B_; ++i) s += part[i];   // fixed order -> deterministic
        out[0] = s / (float)B_;
    }
}

extern "C" void kernel_launch(void* const* d_in, const int* in_sizes, int n_in,
                              void* d_out, int out_size, void* d_ws, size_t ws_size,
                              hipStream_t stream) {
    (void)in_sizes; (void)n_in; (void)out_size; (void)ws_size;
    const float* x     = (const float*)d_in[0];
    const float* iw    = (const float*)d_in[1];
    const float* ib    = (const float*)d_in[2];
    const float* sw    = (const float*)d_in[3];
    const float* smu   = (const float*)d_in[4];
    const float* ssig  = (const float*)d_in[5];
    const float* serev = (const float*)d_in[6];
    const float* w     = (const float*)d_in[7];
    const float* mu    = (const float*)d_in[8];
    const float* sig   = (const float*)d_in[9];
    const float* erev  = (const float*)d_in[10];
    const float* gleak = (const float*)d_in[11];
    const float* vleak = (const float*)d_in[12];
    const float* cm    = (const float*)d_in[13];
    const float* ow    = (const float*)d_in[14];
    const float* ob    = (const float*)d_in[15];
    const float* hw    = (const float*)d_in[16];
    const float* hb    = (const float*)d_in[17];

    float* out = (float*)d_out;
    float* out_seq   = out;                                  // [B,T,U]
    float* out_state = out + (size_t)B_ * T_ * U_;           // [B,U]
    float* out_pond  = out + (size_t)B_ * T_ * U_ + (size_t)B_ * U_;  // [1]

    char* ws = (char*)d_ws;
    float4* pk = (float4*)ws;                                        // U*U float4 = 1 MB
    float4* sk = (float4*)(ws + (size_t)U_ * U_ * sizeof(float4));   // S*U float4 = 512 KB
    float*  pp = (float*)(ws + (size_t)U_ * U_ * sizeof(float4)
                             + (size_t)S_ * U_ * sizeof(float4));    // B floats

    actltc_prep<<<(U_ * U_ + 255) / 256, 256, 0, stream>>>(w, mu, sig, erev, pk, U_ * U_);
    actltc_prep<<<(S_ * U_ + 255) / 256, 256, 0, stream>>>(sw, smu, ssig, serev, sk, S_ * U_);
    actltc_main<<<B_, 512, 0, stream>>>(pk, sk, x, iw, ib, gleak, vleak, cm,
                                        ow, ob, hw, hb, out_seq, out_state, pp);
    actltc_finalize<<<1, 32, 0, stream>>>(pp, out_pond);
}